// KMVGAE_76192719831424
// MI455X (gfx1250) — compile-verified
//
#include <hip/hip_runtime.h>
#include <hip/hip_bf16.h>
#include <math.h>

// ---------------- problem constants (match reference) ----------------
#define NN    8192      // nodes
#define FF    2048      // input features
#define EE    262144    // edges
#define HH1   32
#define HH2   16
#define ALPHA 0.9f

typedef __attribute__((ext_vector_type(2))) float v2f;
typedef __attribute__((ext_vector_type(8))) float v8f;

__device__ __forceinline__ v8f wmma4(v2f a, v2f b, v8f c) {
    // V_WMMA_F32_16X16X4_F32 : D = A(16x4) * B(4x16) + C(16x16)
    return __builtin_amdgcn_wmma_f32_16x16x4_f32(
        /*neg_a=*/false, a, /*neg_b=*/false, b,
        /*c_mod=*/(short)0, c, /*reuse_a=*/false, /*reuse_b=*/false);
}

// ---------------- utility ----------------
__global__ void k_zero(float* __restrict__ p, int count) {
    int i = blockIdx.x * blockDim.x + threadIdx.x;
    if (i < count) p[i] = 0.0f;
}

__global__ void k_degree(const int* __restrict__ dst, float* __restrict__ deg, int e_cnt) {
    int i = blockIdx.x * blockDim.x + threadIdx.x;
    if (i < e_cnt) atomicAdd(&deg[dst[i]], 1.0f);
}

__global__ void k_norm(float* __restrict__ nrm) {   // in place: deg -> deg^{-1/2}
    int i = blockIdx.x * blockDim.x + threadIdx.x;
    if (i < NN) nrm[i] = rsqrtf(fmaxf(nrm[i], 1.0f));
}

// ---------------- GEMM1: m0 = norm * (feat @ W0)   [NN x 32] ----------------
// feat (64 MB stream) is double-buffered through LDS with the gfx1250 async
// copy engine: global_load_async_to_lds_b128 (ASYNCcnt), overlapped with WMMA
// on the previous tile. Block = 4 waves = 64 rows; K-tile = 64.
#define TK      64
#define LDSROW  68          // 64 + 4 pad floats: 272B row stride, 16B aligned, bank-conflict free

__global__ void __launch_bounds__(128)
k_gemm_feat_w0(const float* __restrict__ feat,
               const float* __restrict__ W0,
               const float* __restrict__ nrm,
               float* __restrict__ m0) {
    __shared__ float tile[2][64 * LDSROW];     // 2 x ~17 KB

    const int tid  = threadIdx.x;
    const int lane = tid & 31;
    const int wave = tid >> 5;                 // 0..3
    const int lr   = lane & 15;
    const int half = lane >> 4;
    const int rowBlock = blockIdx.x * 64;      // 64 feat rows per block

    v8f acc0 = {}; v8f acc1 = {};

    // cooperative async fill of one 64x64 f32 tile: 1024 b128 chunks, 8/thread
    auto issue_tile = [&](int buf, int k0) {
#pragma unroll
        for (int i = 0; i < 8; ++i) {
            const int c   = tid + i * 128;     // 0..1023
            const int r   = c >> 4;            // 0..63  (16 b128 chunks per row)
            const int c16 = c & 15;
            const float* g = feat + (size_t)(rowBlock + r) * FF + k0 + c16 * 4;
            const unsigned lds = (unsigned)(size_t)&tile[buf][r * LDSROW + c16 * 4];
            asm volatile("global_load_async_to_lds_b128 %0, %1, off"
                         :: "v"(lds), "v"((unsigned long long)g) : "memory");
        }
    };

    issue_tile(0, 0);
    for (int kt = 0; kt < FF / TK; ++kt) {
        const int buf = kt & 1;
        asm volatile("s_wait_asynccnt 0" ::: "memory");   // this buffer's fills done
        __syncthreads();                                  // visible to all waves
        if (kt + 1 < FF / TK) issue_tile(buf ^ 1, (kt + 1) * TK);

        const float* arow = &tile[buf][(wave * 16 + lr) * LDSROW + 2 * half];
#pragma unroll
        for (int kk = 0; kk < TK; kk += 4) {
            float2 at = *(const float2*)(arow + kk);      // ds_load_b64
            v2f a; a.x = at.x; a.y = at.y;
            const int kr = kt * TK + kk + 2 * half;       // W0 stays L2-resident
            v2f b0; b0.x = W0[(size_t)kr * HH1 + lr];      b0.y = W0[(size_t)(kr + 1) * HH1 + lr];
            v2f b1; b1.x = W0[(size_t)kr * HH1 + 16 + lr]; b1.y = W0[(size_t)(kr + 1) * HH1 + 16 + lr];
            acc0 = wmma4(a, b0, acc0);
            acc1 = wmma4(a, b1, acc1);
        }
        __syncthreads();   // done reading buf before it gets re-targeted
    }

#pragma unroll
    for (int r = 0; r < 8; ++r) {
        const int m = rowBlock + wave * 16 + r + half * 8;  // C/D: VGPR r = row r (+8 upper half)
        const float s = nrm[m];
        m0[(size_t)m * HH1 + lr]      = s * acc0[r];
        m0[(size_t)m * HH1 + 16 + lr] = s * acc1[r];
    }
}

// ---------------- edge scatter: agg[dst] += m0[src]   (32 floats/edge) ----------------
__global__ void k_scatter32(const float* __restrict__ v,
                            const int* __restrict__ src, const int* __restrict__ dst,
                            float* __restrict__ agg, int e_cnt) {
    long t = (long)blockIdx.x * blockDim.x + threadIdx.x;
    int e = (int)(t >> 3), c = (int)(t & 7);
    if (e >= e_cnt) return;
    float4 x = *(const float4*)&v[(size_t)src[e] * HH1 + c * 4];
    float* p = &agg[(size_t)dst[e] * HH1 + c * 4];
    atomicAdd(p + 0, x.x); atomicAdd(p + 1, x.y);
    atomicAdd(p + 2, x.z); atomicAdd(p + 3, x.w);
}

__global__ void k_relu(const float* __restrict__ agg, const float* __restrict__ nrm,
                       float* __restrict__ h) {
    int i = blockIdx.x * blockDim.x + threadIdx.x;
    if (i < NN * HH1) h[i] = fmaxf(nrm[i >> 5] * agg[i], 0.0f);
}

// ---------------- GEMM2: mm = norm * (h @ Wm)   [NN x 16] ----------------
__global__ void k_gemm_h_wm(const float* __restrict__ h,
                            const float* __restrict__ Wm,
                            const float* __restrict__ nrm,
                            float* __restrict__ mm) {
    const int lane = threadIdx.x & 31;
    const int wid  = (blockIdx.x * blockDim.x + threadIdx.x) >> 5;  // 0..511
    const int lr   = lane & 15;
    const int half = lane >> 4;
    const int row  = wid * 16 + lr;
    v8f acc = {};
    const float* arow = h + (size_t)row * HH1 + 2 * half;
#pragma unroll
    for (int k0 = 0; k0 < HH1; k0 += 4) {
        float2 at = *(const float2*)(arow + k0);
        v2f a; a.x = at.x; a.y = at.y;
        const int kr = k0 + 2 * half;
        v2f b; b.x = Wm[kr * HH2 + lr]; b.y = Wm[(kr + 1) * HH2 + lr];
        acc = wmma4(a, b, acc);
    }
#pragma unroll
    for (int r = 0; r < 8; ++r) {
        const int m = wid * 16 + r + half * 8;
        mm[(size_t)m * HH2 + lr] = nrm[m] * acc[r];
    }
}

// ---------------- edge scatter: agg_m[dst] += mm[src]   (16 floats/edge) --------------
__global__ void k_scatter16(const float* __restrict__ v,
                            const int* __restrict__ src, const int* __restrict__ dst,
                            float* __restrict__ agg, int e_cnt) {
    long t = (long)blockIdx.x * blockDim.x + threadIdx.x;
    int e = (int)(t >> 2), c = (int)(t & 3);
    if (e >= e_cnt) return;
    float4 x = *(const float4*)&v[(size_t)src[e] * HH2 + c * 4];
    float* p = &agg[(size_t)dst[e] * HH2 + c * 4];
    atomicAdd(p + 0, x.x); atomicAdd(p + 1, x.y);
    atomicAdd(p + 2, x.z); atomicAdd(p + 3, x.w);
}

// ---------------- edge gate: z_auto[dst] += h[src] * tanh([h_dst,h_src].gw+b)*d_d*d_s*a
__global__ void k_gate(const float* __restrict__ h,
                       const float* __restrict__ gate_w, const float* __restrict__ gate_b,
                       const float* __restrict__ nrm,
                       const int* __restrict__ src, const int* __restrict__ dst,
                       float* __restrict__ z_auto, int e_cnt) {
    __shared__ float s_gw[2 * HH1];
    __shared__ float s_gb;
    if (threadIdx.x < 2 * HH1) s_gw[threadIdx.x] = gate_w[threadIdx.x];
    if (threadIdx.x == 0) s_gb = gate_b[0];
    __syncthreads();
    int e = blockIdx.x * blockDim.x + threadIdx.x;
    if (e >= e_cnt) return;
    const int s = src[e], d = dst[e];
    const float4* hs4 = (const float4*)&h[(size_t)s * HH1];
    const float4* hd4 = (const float4*)&h[(size_t)d * HH1];
    float4 hs[8];
    float acc = s_gb;
#pragma unroll
    for (int c = 0; c < 8; ++c) {
        float4 hd = hd4[c];
        hs[c] = hs4[c];
        acc += hd.x * s_gw[c * 4 + 0] + hd.y * s_gw[c * 4 + 1]
             + hd.z * s_gw[c * 4 + 2] + hd.w * s_gw[c * 4 + 3];
        acc += hs[c].x * s_gw[HH1 + c * 4 + 0] + hs[c].y * s_gw[HH1 + c * 4 + 1]
             + hs[c].z * s_gw[HH1 + c * 4 + 2] + hs[c].w * s_gw[HH1 + c * 4 + 3];
    }
    const float ev = tanhf(acc) * nrm[d] * nrm[s] * ALPHA;
    float* p = &z_auto[(size_t)d * HH1];
#pragma unroll
    for (int c = 0; c < 8; ++c) {
        atomicAdd(p + c * 4 + 0, hs[c].x * ev);
        atomicAdd(p + c * 4 + 1, hs[c].y * ev);
        atomicAdd(p + c * 4 + 2, hs[c].z * ev);
        atomicAdd(p + c * 4 + 3, hs[c].w * ev);
    }
}

// ---------------- logstd = (z_auto + a*h) @ t2_w ; z = noise*exp(logstd) + norm*agg_m
__global__ void k_logstd_z(const float* __restrict__ z_auto, const float* __restrict__ h,
                           const float* __restrict__ t2_w,  const float* __restrict__ noise,
                           const float* __restrict__ nrm,   const float* __restrict__ agg_m,
                           float* __restrict__ z) {
    __shared__ float s_t2[HH1 * HH2];
    for (int i = threadIdx.x; i < HH1 * HH2; i += blockDim.x) s_t2[i] = t2_w[i];
    __syncthreads();
    int n = blockIdx.x * blockDim.x + threadIdx.x;
    if (n >= NN) return;
    float in[HH1];
#pragma unroll
    for (int j = 0; j < HH1; ++j) in[j] = z_auto[(size_t)n * HH1 + j] + ALPHA * h[(size_t)n * HH1 + j];
    const float dn = nrm[n];
#pragma unroll
    for (int c = 0; c < HH2; ++c) {
        float ls = 0.0f;
#pragma unroll
        for (int j = 0; j < HH1; ++j) ls += in[j] * s_t2[j * HH2 + c];
        z[(size_t)n * HH2 + c] = noise[(size_t)n * HH2 + c] * __expf(ls)
                               + dn * agg_m[(size_t)n * HH2 + c];
    }
}

// ---------------- A_pred = sigmoid(z @ z^T)  [NN x NN], WMMA, coalesced stores -------
__global__ void k_zzT_sigmoid(const float* __restrict__ z, float* __restrict__ out) {
    const int lane = threadIdx.x & 31;
    const int wave = threadIdx.x >> 5;                 // 0..3
    const int lr   = lane & 15;
    const int half = lane >> 4;
    const int col0 = (blockIdx.x * 4 + wave) * 16;
    const int row0 = blockIdx.y * 16;
    v8f acc = {};
    const float* za = z + (size_t)(row0 + lr) * HH2 + 2 * half;   // A frag: rows
    const float* zb = z + (size_t)(col0 + lr) * HH2 + 2 * half;   // B frag: cols (same pattern)
#pragma unroll
    for (int k0 = 0; k0 < HH2; k0 += 4) {
        float2 at = *(const float2*)(za + k0);
        float2 bt = *(const float2*)(zb + k0);
        v2f a; a.x = at.x; a.y = at.y;
        v2f b; b.x = bt.x; b.y = bt.y;
        acc = wmma4(a, b, acc);
    }
#pragma unroll
    for (int r = 0; r < 8; ++r) {
        const int m = row0 + r + half * 8;
        out[(size_t)m * NN + col0 + lr] = 1.0f / (1.0f + __expf(-acc[r]));
    }
}

// ---------------- host side ----------------
extern "C" void kernel_launch(void* const* d_in, const int* in_sizes, int n_in,
                              void* d_out, int out_size, void* d_ws, size_t ws_size,
                              hipStream_t stream) {
    const float* feat   = (const float*)d_in[0];
    const float* W0     = (const float*)d_in[1];
    const float* Wm     = (const float*)d_in[2];
    const float* gate_w = (const float*)d_in[3];
    const float* gate_b = (const float*)d_in[4];
    const float* t2_w   = (const float*)d_in[5];
    const float* noise  = (const float*)d_in[6];
    const int*   src    = (const int*)d_in[7];
    const int*   dst    = (const int*)d_in[8];
    float* out = (float*)d_out;
    const int e_cnt = in_sizes[7];   // == EE

    // workspace layout (floats); total = 177*NN ~ 5.8 MB
    float* w      = (float*)d_ws;
    float* nrm    = w;                   // NN      (used first as degree accumulator)
    float* m0     = nrm    + NN;         // NN*32
    float* agg_h  = m0     + NN * HH1;   // NN*32
    float* h      = agg_h  + NN * HH1;   // NN*32
    float* mm     = h      + NN * HH1;   // NN*16
    float* agg_m  = mm     + NN * HH2;   // NN*16
    float* z_auto = agg_m  + NN * HH2;   // NN*32
    float* z      = z_auto + NN * HH1;   // NN*16

    // 1) zero deg/agg buffers (covers nrm..z_auto range = 161*NN floats)
    const int zcnt = NN * 161;
    k_zero<<<(zcnt + 255) / 256, 256, 0, stream>>>(nrm, zcnt);

    // 2-3) degree -> norm
    k_degree<<<(e_cnt + 255) / 256, 256, 0, stream>>>(dst, nrm, e_cnt);
    k_norm<<<(NN + 255) / 256, 256, 0, stream>>>(nrm);

    // 4) m0 = norm * (feat @ W0)     (async-LDS double-buffered, 64 rows/block)
    k_gemm_feat_w0<<<NN / 64, 128, 0, stream>>>(feat, W0, nrm, m0);

    // 5-6) h = relu(norm * segsum(m0[src] -> dst))
    k_scatter32<<<(e_cnt * 8 + 255) / 256, 256, 0, stream>>>(m0, src, dst, agg_h, e_cnt);
    k_relu<<<(NN * HH1 + 255) / 256, 256, 0, stream>>>(agg_h, nrm, h);

    // 7-8) mean path
    k_gemm_h_wm<<<128, 128, 0, stream>>>(h, Wm, nrm, mm);
    k_scatter16<<<(e_cnt * 4 + 255) / 256, 256, 0, stream>>>(mm, src, dst, agg_m, e_cnt);

    // 9) edge-gated aggregation
    k_gate<<<(e_cnt + 255) / 256, 256, 0, stream>>>(h, gate_w, gate_b, nrm, src, dst, z_auto, e_cnt);

    // 10) z = noise*exp(logstd) + mean
    k_logstd_z<<<(NN + 255) / 256, 256, 0, stream>>>(z_auto, h, t2_w, noise, nrm, agg_m, z);

    // 11) A_pred = sigmoid(z @ z^T)   (512x512 tiles; 4 col-tiles per block)
    dim3 grid(NN / (16 * 4), NN / 16);
    k_zzT_sigmoid<<<grid, 128, 0, stream>>>(z, out);
}